// PseudoMambaCore_35631048688101
// MI455X (gfx1250) — compile-verified
//
#include <hip/hip_runtime.h>

// y = g*h + (1-g)*tanh(x),  g = sigmoid(x)
// Memory-bound streaming kernel: 12 B/elem @ 23.3 TB/s HBM.
// Strategy: float4 (b128) non-temporal loads/stores, one v_exp_f32 per element.

typedef __attribute__((ext_vector_type(4))) float v4f;

#define LOG2E 1.44269504088896340736f

__device__ __forceinline__ float gate_blend(float x, float h) {
  // e = 2^(-x*log2e) = exp(-x)
  float e  = __builtin_amdgcn_exp2f(-LOG2E * x);
  // sigmoid(x) = 1/(1+e)
  float g  = __builtin_amdgcn_rcpf(1.0f + e);
  // tanh(x) = 2/(1+e^2) - 1   (reuses e; saturates correctly: e=inf -> t=-1)
  float e2 = e * e;
  float t  = __builtin_fmaf(2.0f, __builtin_amdgcn_rcpf(1.0f + e2), -1.0f);
  // y = g*h + (1-g)*t = g*(h-t) + t
  return __builtin_fmaf(g, h - t, t);
}

__global__ __launch_bounds__(256) void pseudo_mamba_core_kernel(
    const float* __restrict__ x, const float* __restrict__ h,
    float* __restrict__ y, long long n4, long long n) {
  const long long stride = (long long)gridDim.x * blockDim.x;
  long long i = (long long)blockIdx.x * blockDim.x + threadIdx.x;

  const v4f* __restrict__ x4 = (const v4f*)x;
  const v4f* __restrict__ h4 = (const v4f*)h;
  v4f* __restrict__ y4 = (v4f*)y;

  for (; i < n4; i += stride) {
    // Non-temporal 128-bit loads: streaming data, never reused -> don't
    // pollute WGP$/L2 (working set >> 192MB L2).
    v4f xv = __builtin_nontemporal_load(x4 + i);
    v4f hv = __builtin_nontemporal_load(h4 + i);
    v4f yv;
    yv.x = gate_blend(xv.x, hv.x);
    yv.y = gate_blend(xv.y, hv.y);
    yv.z = gate_blend(xv.z, hv.z);
    yv.w = gate_blend(xv.w, hv.w);
    __builtin_nontemporal_store(yv, y4 + i);
  }

  // Scalar tail for n not divisible by 4 (not hit for 8192*4096, kept for safety).
  long long tail = n4 * 4 + ((long long)blockIdx.x * blockDim.x + threadIdx.x);
  if (tail < n) {
    y[tail] = gate_blend(x[tail], h[tail]);
  }
}

extern "C" void kernel_launch(void* const* d_in, const int* in_sizes, int n_in,
                              void* d_out, int out_size, void* d_ws, size_t ws_size,
                              hipStream_t stream) {
  const float* x = (const float*)d_in[0];
  const float* h = (const float*)d_in[1];
  float* y = (float*)d_out;

  const long long n  = (long long)in_sizes[0];   // 8192*4096 = 33,554,432
  const long long n4 = n >> 2;                   // 8,388,608 float4 items

  const int block = 256;                         // 8 wave32s per block
  long long grid_ll = (n4 + block - 1) / block;  // 32768 blocks: covers all in one pass
  if (grid_ll < 1) grid_ll = 1;
  const unsigned grid = (unsigned)grid_ll;

  pseudo_mamba_core_kernel<<<grid, block, 0, stream>>>(x, h, y, n4, n);
}